// LRInvalidation_21844203668018
// MI455X (gfx1250) — compile-verified
//
#include <hip/hip_runtime.h>
#include <hip/hip_bf16.h>

// LR-consistency invalidation mask.
// dl, dr: [N*H, W] fp32 (N=16,C=1,H=720,W=1280). out: fp32 0/1 mask.
// One workgroup (320 threads = W/4) per image row:
//   1) each lane issues ONE global_load_async_to_lds_b128 to stage its 16B
//      slice of the dr row into LDS (ASYNCcnt-tracked CDNA5 DMA),
//   2) overlapped float4 load of this lane's 4 dl pixels,
//   3) s_wait_asynccnt 0 + barrier,
//   4) horizontal bilinear gather from LDS, float4 store of the 0/1 mask.
// All global traffic is single-touch, 16B/lane, coalesced: ~177MB total
// => ~7.6us at 23.3 TB/s HBM. Pure bandwidth kernel; no matmul => no WMMA.

#define W_DIM   1280
#define BLOCK   (W_DIM / 4)      // 320 threads, 10 wave32s
#define LAMBDA  0.6931f

// Builtin signature (probe-confirmed): (AS1 int4* src, AS3 int4* dst, imm, imm).
typedef int v4i __attribute__((ext_vector_type(4)));
typedef __attribute__((address_space(1))) v4i* gv4i_t;
typedef __attribute__((address_space(3))) v4i* lv4i_t;

__global__ __launch_bounds__(BLOCK)
void lr_invalidation_kernel(const float* __restrict__ dl,
                            const float* __restrict__ dr,
                            float* __restrict__ out)
{
    __shared__ __align__(16) float s_dr[W_DIM];

    const int  tid  = threadIdx.x;
    const long long base = (long long)blockIdx.x * W_DIM;
    const int  c4   = 4 * tid;          // first pixel of this lane's float4

    // ---- 1) kick off async DMA of the dr row into LDS (one b128 per lane) ----
#if defined(__gfx1250__) && __has_builtin(__builtin_amdgcn_global_load_async_to_lds_b128)
    __builtin_amdgcn_global_load_async_to_lds_b128(
        (gv4i_t)(dr + base + c4),
        (lv4i_t)(&s_dr[c4]),
        /*offset=*/0, /*cpol=*/0);
#else
    for (int k = tid; k < W_DIM; k += BLOCK) s_dr[k] = dr[base + k];
#endif

    // ---- 2) overlap: vector load of this lane's 4 dl pixels ----
    const float4 dl4 = *reinterpret_cast<const float4*>(dl + base + c4);

    // ---- 3) wait for DMA completion, then workgroup barrier ----
#if defined(__gfx1250__) && __has_builtin(__builtin_amdgcn_global_load_async_to_lds_b128)
  #if __has_builtin(__builtin_amdgcn_s_wait_asynccnt)
    __builtin_amdgcn_s_wait_asynccnt(0);
  #else
    asm volatile("s_wait_asynccnt 0" ::: "memory");
  #endif
#endif
    __syncthreads();

    // ---- 4) compute 4 pixels, store as one float4 ----
    const float dlv_arr[4] = { dl4.x, dl4.y, dl4.z, dl4.w };
    float r[4];

    #pragma unroll
    for (int p = 0; p < 4; ++p) {
        const int   j   = c4 + p;
        const float dlv = dlv_arr[p];

        const float x   = (float)j - dlv;
        const float x0f = floorf(x);
        const float wx  = x - x0f;
        const int   x0  = (int)x0f;
        const int   x1  = x0 + 1;

        const float v0 = (x0 >= 0 && x0 < W_DIM) ? 1.0f : 0.0f;
        const float v1 = (x1 >= 0 && x1 < W_DIM) ? 1.0f : 0.0f;
        const int x0c = min(max(x0, 0), W_DIM - 1);
        const int x1c = min(max(x1, 0), W_DIM - 1);

        const float g0 = s_dr[x0c];
        const float g1 = s_dr[x1c];

        const float recon = (1.0f - wx) * g0 * v0 + wx * g1 * v1;
        const float diff  = fabsf(dlv - recon);
        const float prob  = expf(-LAMBDA * diff);

        r[p] = (prob < 0.5f) ? 1.0f : 0.0f;
    }

    *reinterpret_cast<float4*>(out + base + c4) =
        make_float4(r[0], r[1], r[2], r[3]);
}

extern "C" void kernel_launch(void* const* d_in, const int* in_sizes, int n_in,
                              void* d_out, int out_size, void* d_ws, size_t ws_size,
                              hipStream_t stream)
{
    const float* dl = (const float*)d_in[0];
    const float* dr = (const float*)d_in[1];
    float* out      = (float*)d_out;

    const int rows = in_sizes[0] / W_DIM;   // N*C*H = 11520

    lr_invalidation_kernel<<<dim3(rows), dim3(BLOCK), 0, stream>>>(dl, dr, out);
}